// SoftCluster_57088705298485
// MI455X (gfx1250) — compile-verified
//
#include <hip/hip_runtime.h>

// SoftCluster fused kernel for MI455X (gfx1250, wave32).
//
//   xt[b,n,d] = sum_t fc_w[t] * x[b,t,n,d]            (streamed, 201 MB read)
//   z[b,c,n]  = sum_d row_W[c,d] * xt[b,n,d] + fc_b   (V_WMMA_F32_16X16X4_F32)
//   out       = softmax over c                         (134 MB write)
//
// Memory-bound: ~336 MB @ 23.3 TB/s => ~14.4 us floor. One pass over x, one
// pass over out; xt tile lives in LDS.

typedef __attribute__((ext_vector_type(2))) float v2f;
typedef __attribute__((ext_vector_type(8))) float v8f;

#define B_ 64
#define T_ 64
#define N_ 4096
#define D_ 3
#define C_ 128

#define NODES_PER_BLOCK 128
#define THREADS 256

__global__ __launch_bounds__(THREADS)
void SoftCluster_57088705298485_kernel(const float* __restrict__ x,
                                       const float* __restrict__ row_W,
                                       const float* __restrict__ fc_w,
                                       const float* __restrict__ fc_b,
                                       float* __restrict__ out) {
    __shared__ float xt[NODES_PER_BLOCK * D_];  // 384 floats = 1.5 KB

    const int tid = threadIdx.x;
    const int n0  = blockIdx.x * NODES_PER_BLOCK;
    const int b   = blockIdx.y;

    // ---- Phase 1: hist contraction (fully coalesced streaming of x) ----
    // flat index f in [0,384) maps to node n = f/3, dim d = f%3.
    // Thread tid owns f = tid and (tid < 128) f = 256 + tid.
    const float* xb = x + (size_t)b * T_ * (N_ * D_) + (size_t)n0 * D_;
    float acc0 = 0.f, acc1 = 0.f;
#pragma unroll 4
    for (int t = 0; t < T_; ++t) {
        const float w = fc_w[t];                       // uniform -> s_load
        const float* xr = xb + (size_t)t * (N_ * D_);  // 1536 B contiguous row
        acc0 += w * xr[tid];
        if (tid < NODES_PER_BLOCK * D_ - THREADS)      // waves 0-3 only (uniform)
            acc1 += w * xr[THREADS + tid];
    }
    xt[tid] = acc0;
    if (tid < NODES_PER_BLOCK * D_ - THREADS) xt[THREADS + tid] = acc1;
    __syncthreads();

    // ---- Phase 2: cluster GEMM via WMMA + softmax ----
    const int lane = tid & 31;
    const int wv   = tid >> 5;        // wave 0..7 -> nodes [n0+16*wv, +16)
    const int nl   = lane & 15;
    const bool hi  = lane >= 16;

    // B fragment (4x16 f32, K x node): lanes 0-15 carry K=0,1; lanes 16-31 K=2,3.
    // K=3 padded with zero (D=3).
    const float* xg = &xt[(wv * 16 + nl) * D_];
    const float e0 = xg[0], e1 = xg[1], e2 = xg[2];
    v2f bfrag;
    bfrag.x = hi ? e2 : e0;
    bfrag.y = hi ? 0.f : e1;

    // C fragment: broadcast bias (constant over c => cancels in softmax, but free).
    const float fcb = fc_b[0];
    v8f cfrag;
#pragma unroll
    for (int r = 0; r < 8; ++r) cfrag[r] = fcb;

    // A fragments (16x4 f32, cluster x K) from row_W; 8 tiles cover C=128.
    v8f acc[8];
#pragma unroll
    for (int j = 0; j < 8; ++j) {
        const float* rw = row_W + (j * 16 + nl) * D_;
        const float r0 = rw[0], r1 = rw[1], r2 = rw[2];
        v2f afrag;
        afrag.x = hi ? r2 : r0;
        afrag.y = hi ? 0.f : r1;
        // z_tile[16c x 16n] = A x B + bias   (EXEC all ones here)
        acc[j] = __builtin_amdgcn_wmma_f32_16x16x4_f32(
            false, afrag, false, bfrag, (short)0, cfrag, false, false);
    }

    // Softmax over c=128 for node n = n0+16*wv+nl.
    // This lane holds 64 c-values; partner lane (lane^16) holds the other 64.
    float m = -__builtin_huge_valf();
#pragma unroll
    for (int j = 0; j < 8; ++j)
#pragma unroll
        for (int r = 0; r < 8; ++r) m = fmaxf(m, acc[j][r]);
    m = fmaxf(m, __shfl_xor(m, 16, 32));

    float s = 0.f;
#pragma unroll
    for (int j = 0; j < 8; ++j)
#pragma unroll
        for (int r = 0; r < 8; ++r) {
            const float e = __expf(acc[j][r] - m);
            acc[j][r] = e;
            s += e;
        }
    s += __shfl_xor(s, 16, 32);
    const float inv = 1.0f / s;

    // Store: D layout => for fixed (j,r), lanes 0-15 write 16 consecutive n at
    // row c=j*16+r, lanes 16-31 at row c=j*16+8+r (two 64B segments per store).
    const int n = n0 + wv * 16 + nl;
    float* ob = out + (size_t)b * C_ * N_ + n;
    const int chi = hi ? 8 : 0;
#pragma unroll
    for (int j = 0; j < 8; ++j)
#pragma unroll
        for (int r = 0; r < 8; ++r) {
            const int c = j * 16 + chi + r;
            ob[(size_t)c * N_] = acc[j][r] * inv;
        }
}

extern "C" void kernel_launch(void* const* d_in, const int* in_sizes, int n_in,
                              void* d_out, int out_size, void* d_ws, size_t ws_size,
                              hipStream_t stream) {
    const float* x     = (const float*)d_in[0];   // [B,T,N,D]
    // d_in[1] = orinode_loc, unused by the reference computation
    const float* row_W = (const float*)d_in[2];   // [C,D]
    const float* fc_w  = (const float*)d_in[3];   // [T]
    const float* fc_b  = (const float*)d_in[4];   // [1]
    float* out = (float*)d_out;                   // [B,C,N]

    dim3 grid(N_ / NODES_PER_BLOCK, B_);          // (32, 64)
    dim3 block(THREADS);
    SoftCluster_57088705298485_kernel<<<grid, block, 0, stream>>>(
        x, row_W, fc_w, fc_b, out);
}